// TorchFeedForwardNetwork_63041529970912
// MI455X (gfx1250) — compile-verified
//
#include <hip/hip_runtime.h>
#include <hip/hip_bf16.h>
#include <cstdint>

#define N_IN   128
#define LAYERS 8
#define WIDTH  256
#define KFAN   16
#define ROWS   32                              // batch rows per workgroup (lane == row)
#define F_TOT  (N_IN + LAYERS * WIDTH)         // 2176 features total
#define LDS_BYTES (F_TOT * ROWS * 4)           // 278528 B — needs CDNA5's 320KB WGP LDS

#if defined(__has_builtin)
# if __has_builtin(__builtin_amdgcn_global_load_async_to_lds_b32)
#  define HAVE_ASYNC_LD 1
# endif
# if __has_builtin(__builtin_amdgcn_global_store_async_from_lds_b32)
#  define HAVE_ASYNC_ST 1
# endif
# if __has_builtin(__builtin_amdgcn_s_wait_asynccnt)
#  define HAVE_WAIT_ASYNC 1
# endif
#endif

typedef __attribute__((address_space(1))) int gint_t;
typedef __attribute__((address_space(3))) int lint_t;

__device__ __forceinline__ void async_copy_in_b32(const float* g, float* l) {
#if defined(HAVE_ASYNC_LD)
  // gfx1250 async DMA: per-lane global src, per-lane LDS dst (transpose scatter),
  // tracked on ASYNCcnt, overlapped with subsequent work.
  __builtin_amdgcn_global_load_async_to_lds_b32((gint_t*)(g), (lint_t*)(l), 0, 0);
#else
  *l = *g;  // fallback: plain load + ds_store
#endif
}

__device__ __forceinline__ void async_copy_out_b32(float* g, const float* l) {
#if defined(HAVE_ASYNC_ST)
  // LDS -> global via async engine: bank conflicts on the strided LDS read are
  // absorbed off the wave's critical path; global side stays coalesced.
  __builtin_amdgcn_global_store_async_from_lds_b32((gint_t*)(g), (lint_t*)(l), 0, 0);
#else
  *g = *l;
#endif
}

__device__ __forceinline__ void wait_async0() {
#if defined(HAVE_WAIT_ASYNC)
  __builtin_amdgcn_s_wait_asynccnt(0);
#elif defined(HAVE_ASYNC_LD) || defined(HAVE_ASYNC_ST)
  asm volatile("s_wait_asynccnt 0" ::: "memory");
#endif
}

__global__ void __launch_bounds__(256)
ffn_sparse_kernel(const float* __restrict__ inputs,
                  const float* __restrict__ weights,
                  const float* __restrict__ biases,
                  const int*   __restrict__ edge_idx,
                  float* __restrict__ out) {
  extern __shared__ float vals[];              // layout: vals[f * ROWS + row]
  const int t    = threadIdx.x;
  const int lane = t & 31;                     // lane == batch row in tile
  const int wv   = __builtin_amdgcn_readfirstlane(t >> 5);  // wave id, SGPR-uniform
  const int row0 = blockIdx.x * ROWS;

  // ---- Stage input tile transposed into LDS: vals[f][r] = inputs[row0+r][f] ----
  // Global side coalesced (consecutive lanes -> consecutive f); LDS-side scatter
  // handled by the async engine.
  {
    const int f  = t & (N_IN - 1);
    const int rh = t >> 7;                     // 0..1 (256 threads / 128 features)
    const float* g = inputs + (size_t)row0 * N_IN + f;
#pragma unroll
    for (int i = 0; i < 16; ++i) {
      const int r = rh * 16 + i;
      async_copy_in_b32(g + (size_t)r * N_IN, &vals[f * ROWS + r]);
    }
  }
  wait_async0();
  __syncthreads();

  // ---- 8 layers: each wave computes 32 of the 256 new neurons ----
  int fbase = N_IN;
  for (int l = 0; l < LAYERS; ++l) {
#pragma unroll 2
    for (int wi = 0; wi < 32; ++wi) {
      const int w    = (wv << 5) + wi;                 // wave-uniform neuron id
      const int base = (l * WIDTH + w) * KFAN;         // uniform -> s_load path
      const int*   idx = edge_idx + base;
      const float* wt  = weights  + base;
      float acc = biases[l * WIDTH + w];
#pragma unroll
      for (int k = 0; k < KFAN; ++k) {
        const int f = idx[k];                          // wave-uniform feature index
        // 32 lanes read 32 consecutive LDS words: conflict-free gather
        acc = __builtin_fmaf(vals[f * ROWS + lane], wt[k], acc);
      }
      // sigmoid(x) = 1 / (1 + 2^(-x*log2e))  -> v_exp_f32 + v_rcp_f32
      const float e = __builtin_amdgcn_exp2f(-1.44269504088896340736f * acc);
      const float s = __builtin_amdgcn_rcpf(1.0f + e);
      vals[(fbase + w) * ROWS + lane] = s;
    }
    fbase += WIDTH;
    __syncthreads();                                   // next layer gathers our outputs
  }

  // ---- Write final WIDTH features: async LDS -> global, coalesced global side ----
  {
    const int obase = F_TOT - WIDTH;                   // 1920
    float* og = out + (size_t)row0 * WIDTH;
#pragma unroll
    for (int i = 0; i < (ROWS * WIDTH) / 256; ++i) {   // 32 iterations
      const int flat = t + i * 256;
      const int r = flat >> 8;                         // / WIDTH
      const int w = flat & (WIDTH - 1);
      async_copy_out_b32(og + flat, &vals[(obase + w) * ROWS + r]);
    }
  }
  wait_async0();                                       // drain before wave exit
}

extern "C" void kernel_launch(void* const* d_in, const int* in_sizes, int n_in,
                              void* d_out, int out_size, void* d_ws, size_t ws_size,
                              hipStream_t stream) {
  const float* inputs   = (const float*)d_in[0];   // (B, 128) f32
  const float* weights  = (const float*)d_in[1];   // (8, 256, 16) f32
  const float* biases   = (const float*)d_in[2];   // (8, 256) f32
  const int*   edge_idx = (const int*)  d_in[3];   // (8, 256, 16) i32
  float* out = (float*)d_out;                      // (B, 256) f32

  const int B = in_sizes[0] / N_IN;                // 32768

  // Opt in to >64KB dynamic LDS (CDNA5 allows up to 320KB per workgroup).
  (void)hipFuncSetAttribute((const void*)ffn_sparse_kernel,
                            hipFuncAttributeMaxDynamicSharedMemorySize, LDS_BYTES);

  dim3 grid(B / ROWS);                             // 1024 workgroups
  dim3 block(256);                                 // 8 wave32s per WGP
  ffn_sparse_kernel<<<grid, block, LDS_BYTES, stream>>>(inputs, weights, biases,
                                                        edge_idx, out);
}